// MolKGNNNet_36498632081408
// MI455X (gfx1250) — compile-verified
//
#include <hip/hip_runtime.h>
#include <hip/hip_bf16.h>
#include <math.h>

// ---------------------------------------------------------------------------
// MolKGNN forward, algebraically collapsed:
//   r_pre[n,j] = sum_i xb[n,i]*M1[i,j] + msg[n]*we[j] + c[j]   (M1 = 5x5)
//   r = swish(r_pre) @ W2 + b2 ;  out = segment_sum(r, batch)
// BatchNorm folded into M1/c and edge affine.  Per-node 5->5->5 transform is
// done 16 nodes/wave with exact-f32 V_WMMA_F32_16X16X4_F32 (2 K-tiles each).
// ---------------------------------------------------------------------------

typedef __attribute__((ext_vector_type(2))) float v2f;
typedef __attribute__((ext_vector_type(8))) float v8f;

// workspace layout (floats)
#define WS_STATS 0    // [0..4] sum_x, [5..9] sumsq_x, [10] sum_e, [11] sumsq_e
#define WS_FOLD  16   // [0..24] M1s rows 0..4, [25..29] we (row 5), [30..34] c,
                      // [35] s_e, [36] t_e
#define WS_MSG   64   // n_nodes floats

__device__ __forceinline__ v8f wmma_f32_16x16x4(v2f a, v2f b, v8f c) {
  // 8-arg pattern: (neg_a, A, neg_b, B, c_mod, C, reuse_a, reuse_b)
  return __builtin_amdgcn_wmma_f32_16x16x4_f32(false, a, false, b,
                                               (short)0, c, false, false);
}

// swish with hardware rcp (v_rcp_f32, ~1 ulp) instead of IEEE div sequence
__device__ __forceinline__ float swish_fast(float v) {
  return v * __builtin_amdgcn_rcpf(1.f + __expf(-v));
}

// ---------------------------------------------------------------------------
__global__ void zero_kernel(float* __restrict__ ws, float* __restrict__ out,
                            int n_nodes, int out_n) {
  int i = blockIdx.x * blockDim.x + threadIdx.x;
  int stride = gridDim.x * blockDim.x;
  for (int k = i; k < n_nodes; k += stride) ws[WS_MSG + k] = 0.f;
  for (int k = i; k < out_n;   k += stride) out[k] = 0.f;
  if (i < 12) ws[WS_STATS + i] = 0.f;
}

// ---------------------------------------------------------------------------
__global__ void stats_kernel(const float* __restrict__ x,
                             const float* __restrict__ e,
                             float* __restrict__ ws,
                             int n_nodes, int n_edges) {
  int tid = blockIdx.x * blockDim.x + threadIdx.x;
  int stride = gridDim.x * blockDim.x;
  float sx[5] = {0,0,0,0,0}, sxx[5] = {0,0,0,0,0};
  for (int n = tid; n < n_nodes; n += stride) {
    const float* xr = x + (size_t)n * 5;
#pragma unroll
    for (int c = 0; c < 5; ++c) { float v = xr[c]; sx[c] += v; sxx[c] += v*v; }
  }
  float se = 0.f, see = 0.f;
  for (int k = tid; k < n_edges; k += stride) { float v = e[k]; se += v; see += v*v; }
  // wave32 reduction
#pragma unroll
  for (int off = 16; off > 0; off >>= 1) {
#pragma unroll
    for (int c = 0; c < 5; ++c) {
      sx[c]  += __shfl_xor(sx[c],  off, 32);
      sxx[c] += __shfl_xor(sxx[c], off, 32);
    }
    se  += __shfl_xor(se,  off, 32);
    see += __shfl_xor(see, off, 32);
  }
  if ((threadIdx.x & 31) == 0) {
#pragma unroll
    for (int c = 0; c < 5; ++c) {
      atomicAdd(&ws[WS_STATS + c],     sx[c]);
      atomicAdd(&ws[WS_STATS + 5 + c], sxx[c]);
    }
    atomicAdd(&ws[WS_STATS + 10], se);
    atomicAdd(&ws[WS_STATS + 11], see);
  }
}

// ---------------------------------------------------------------------------
// Single block of 32 threads: builds folded constants in ws[WS_FOLD..].
__global__ void fold_kernel(float* __restrict__ ws,
                            const float* __restrict__ gx, const float* __restrict__ bx,
                            const float* __restrict__ ge, const float* __restrict__ be,
                            const float* __restrict__ W_gnn, const float* __restrict__ b_gnn,
                            const float* __restrict__ W_e,  const float* __restrict__ W1,
                            const float* __restrict__ b1,
                            int hidden, int n_nodes, int n_edges) {
  __shared__ float P[25];   // (W_gnn @ W1)[i][j]
  __shared__ float sxv[5], txv[5];
  int t = threadIdx.x;
  if (t < 25) {
    int i = t / 5, j = t % 5;
    float acc = 0.f;
    for (int k = 0; k < hidden; ++k) acc += W_gnn[i * hidden + k] * W1[k * 5 + j];
    P[t] = acc;
  }
  if (t < 5) {
    float inv_n = 1.f / (float)n_nodes;
    float mu  = ws[WS_STATS + t] * inv_n;
    float var = ws[WS_STATS + 5 + t] * inv_n - mu * mu;
    float s = gx[t] * rsqrtf(var + 1e-5f);
    sxv[t] = s;
    txv[t] = bx[t] - mu * s;
  }
  if (t == 31) {
    float inv_e = 1.f / (float)n_edges;
    float mu  = ws[WS_STATS + 10] * inv_e;
    float var = ws[WS_STATS + 11] * inv_e - mu * mu;
    float s = ge[0] * rsqrtf(var + 1e-5f);
    ws[WS_FOLD + 35] = s;
    ws[WS_FOLD + 36] = be[0] - mu * s;
  }
  __syncthreads();
  if (t < 25) ws[WS_FOLD + t] = sxv[t / 5] * P[t];      // M1s rows 0..4
  if (t < 5) {
    float we = 0.f, c = 0.f;
    for (int k = 0; k < hidden; ++k) {
      we += W_e[k]   * W1[k * 5 + t];
      c  += b_gnn[k] * W1[k * 5 + t];
    }
    for (int i = 0; i < 5; ++i) c += txv[i] * P[i * 5 + t];
    c += b1[t];
    ws[WS_FOLD + 25 + t] = we;   // "row 5" weight for the msg feature
    ws[WS_FOLD + 30 + t] = c;    // fused bias
  }
}

// ---------------------------------------------------------------------------
__global__ void edge_kernel(const float* __restrict__ e,
                            const int* __restrict__ edge_index,
                            float* __restrict__ ws, int n_edges) {
  float se = ws[WS_FOLD + 35];
  float te = ws[WS_FOLD + 36];
  int i0 = blockIdx.x * blockDim.x + threadIdx.x;
  int stride = gridDim.x * blockDim.x;
  for (int i = i0; i < n_edges; i += stride) {
    int dst = edge_index[n_edges + i];          // edge_index[1][i]
    atomicAdd(&ws[WS_MSG + dst], e[i] * se + te);
  }
}

// ---------------------------------------------------------------------------
// 16 nodes per wave, 4 waves per block.  Computes transposed products so the
// first WMMA's D (row=feature, col=node) feeds the second WMMA's B with just
// two cross-lane shuffles.
__global__ void __launch_bounds__(128)
node_kernel(const float* __restrict__ x,
            const int*   __restrict__ batch,
            const float* __restrict__ W2, const float* __restrict__ b2,
            const float* __restrict__ F,     // ws + WS_FOLD
            const float* __restrict__ msg,   // ws + WS_MSG
            float* __restrict__ out) {
  const int lane = threadIdx.x & 31;
  const int wave = threadIdx.x >> 5;
  const int tile = blockIdx.x * 4 + wave;
  const int j  = lane & 15;     // A-matrix row (out feature) / B-matrix col (node)
  const int kg = lane >> 4;     // K group: lanes 0-15 -> K{0,1}, 16-31 -> K{2,3}
  const bool jok = (j < 5);

  // A operands (weights, transposed): A[M=j][K=i] = W[i][j]
  v2f a0, a1, w0, w1;
  a0.x = jok ? F[(2*kg + 0) * 5 + j] : 0.f;
  a0.y = jok ? F[(2*kg + 1) * 5 + j] : 0.f;
  a1.x = (jok && kg == 0) ? F[4 * 5 + j] : 0.f;   // K=4: x4 weight
  a1.y = (jok && kg == 0) ? F[5 * 5 + j] : 0.f;   // K=5: msg weight
  w0.x = jok ? W2[(2*kg + 0) * 5 + j] : 0.f;
  w0.y = jok ? W2[(2*kg + 1) * 5 + j] : 0.f;
  w1.x = (jok && kg == 0) ? W2[4 * 5 + j] : 0.f;
  w1.y = 0.f;

  // C inits: per-row (M) bias broadcast; rows >=5 and lanes>=16 (M>=8) are 0
  v8f c0, c2;
#pragma unroll
  for (int i = 0; i < 8; ++i) {
    c0[i] = (lane < 16 && i < 5) ? F[30 + i] : 0.f;
    c2[i] = (lane < 16 && i < 5) ? b2[i] : 0.f;
  }

  const int  base = tile * 16;
  const int  nn   = base + j;
  const float* xr = x + (size_t)nn * 5;

  // B operand = X^T tile: VGPR v, lane L -> element (K=2*(L/16)+v, N=L%16)
  v2f b0, b1v;
  b0.x = xr[2*kg + 0];
  b0.y = xr[2*kg + 1];
  float x4 = xr[4];
  float mg = msg[nn];
  b1v.x = kg ? 0.f : x4;     // K=4
  b1v.y = kg ? 0.f : mg;     // K=5 (msg as 6th input feature)

  v8f d = c0;
  d = wmma_f32_16x16x4(a0, b0,  d);   // K = 0..3
  d = wmma_f32_16x16x4(a1, b1v, d);   // K = 4..5 (rest zero-padded)

  // swish on feature rows 0..4 (other rows are exactly 0)
#pragma unroll
  for (int i = 0; i < 5; ++i) d[i] = swish_fast(d[i]);

  // Repack D (row=feat, col=node) into B layout for the second WMMA:
  // lanes 0-15 need rows {0,1}; lanes 16-31 need rows {2,3} from lanes 0-15.
  float d2l = __shfl(d[2], j, 32);
  float d3l = __shfl(d[3], j, 32);
  v2f s0, s1;
  s0.x = (lane < 16) ? d[0] : d2l;
  s0.y = (lane < 16) ? d[1] : d3l;
  s1.x = (lane < 16) ? d[4] : 0.f;   // K=4
  s1.y = 0.f;

  v8f r = c2;
  r = wmma_f32_16x16x4(w0, s0, r);
  r = wmma_f32_16x16x4(w1, s1, r);

  // r: VGPR i, lanes 0-15 -> element (feat i, node lane). Pool by graph id.
  if (lane < 16) {
    int g = batch[base + lane];
    float* op = out + (size_t)g * 5;
    atomicAdd(op + 0, r[0]);
    atomicAdd(op + 1, r[1]);
    atomicAdd(op + 2, r[2]);
    atomicAdd(op + 3, r[3]);
    atomicAdd(op + 4, r[4]);
  }
}

// ---------------------------------------------------------------------------
extern "C" void kernel_launch(void* const* d_in, const int* in_sizes, int n_in,
                              void* d_out, int out_size, void* d_ws, size_t ws_size,
                              hipStream_t stream) {
  const float* x      = (const float*)d_in[0];
  const float* ea     = (const float*)d_in[1];
  const int*   batch  = (const int*)  d_in[2];
  const int*   eidx   = (const int*)  d_in[3];
  const float* gx     = (const float*)d_in[4];
  const float* bx     = (const float*)d_in[5];
  const float* ge     = (const float*)d_in[6];
  const float* be     = (const float*)d_in[7];
  const float* W_gnn  = (const float*)d_in[8];
  const float* b_gnn  = (const float*)d_in[9];
  const float* W_e    = (const float*)d_in[10];
  const float* W1     = (const float*)d_in[11];
  const float* b1     = (const float*)d_in[12];
  const float* W2     = (const float*)d_in[13];
  const float* b2     = (const float*)d_in[14];
  float* out = (float*)d_out;
  float* ws  = (float*)d_ws;

  const int n_nodes = in_sizes[2];      // batch length = 1,000,000
  const int n_edges = in_sizes[1];      // 2,000,000
  const int hidden  = in_sizes[9];      // 110

  zero_kernel <<<2048, 256, 0, stream>>>(ws, out, n_nodes, out_size);
  stats_kernel<<<1024, 256, 0, stream>>>(x, ea, ws, n_nodes, n_edges);
  fold_kernel <<<1, 32, 0, stream>>>(ws, gx, bx, ge, be, W_gnn, b_gnn,
                                     W_e, W1, b1, hidden, n_nodes, n_edges);
  edge_kernel <<<2048, 256, 0, stream>>>(ea, eidx, ws, n_edges);

  const int n_tiles = n_nodes / 16;     // 62500, exact
  node_kernel<<<n_tiles / 4, 128, 0, stream>>>(x, batch, W2, b2,
                                               ws + WS_FOLD, ws + WS_MSG, out);
}